// GATConvEncoder_27565100106035
// MI455X (gfx1250) — compile-verified
//
#include <hip/hip_runtime.h>
#include <hip/hip_bf16.h>

typedef __attribute__((ext_vector_type(2))) float v2f;
typedef __attribute__((ext_vector_type(8))) float v8f;

#define HC   256   // H*C
#define NH   4     // heads
#define CCH  64    // channels per head
#define KIN  300   // n_in
#define NOUTD 768  // nout

// ---------- ordered-float encoding for atomicMax on float ----------
__device__ __forceinline__ unsigned fenc(float f) {
    unsigned u = __float_as_uint(f);
    return (u & 0x80000000u) ? ~u : (u | 0x80000000u);
}
__device__ __forceinline__ float fdec(unsigned u) {
    unsigned b = (u & 0x80000000u) ? (u & 0x7FFFFFFFu) : ~u;
    return __uint_as_float(b);
}

// ---------- workspace init (deterministic per call) ----------
__global__ void init_ws(unsigned* emax, float* denom, float* agg,
                        float* pooled, float* cnt, int n, int g) {
    size_t stride = (size_t)gridDim.x * blockDim.x;
    size_t t = (size_t)blockIdx.x * blockDim.x + threadIdx.x;
    size_t total = (size_t)n * HC;
    const unsigned ninf = 0x007FFFFFu;  // fenc(-inf)
    for (size_t i = t; i < total; i += stride) {
        agg[i] = 0.0f;
        if (i < (size_t)n * NH) { emax[i] = ninf; denom[i] = 0.0f; }
        if (i < (size_t)g * HC) pooled[i] = 0.0f;
        if (i < (size_t)g)      cnt[i] = 0.0f;
    }
}

// ---------- GEMM1: h[N,256] = x[N,300] @ lin_w[300,256] ----------
// One wave handles a 16x32 output slab (two 16x16 WMMA tiles sharing the A fragment).
__global__ __launch_bounds__(32) void gemm_x_linw(const float* __restrict__ x,
                                                  const float* __restrict__ w,
                                                  float* __restrict__ h, int n) {
    int mTile = blockIdx.x;
    int lane = threadIdx.x & 31;
    int mr = lane & 15, half = lane >> 4;
    int row = mTile * 16 + mr;
    if (row >= n) row = n - 1;              // clamp (N is a multiple of 16 anyway)
    int col0 = blockIdx.y * 32 + mr;        // first tile's column for this lane
    const float* arow = x + (size_t)row * KIN;
    v8f acc0 = {}, acc1 = {};
    for (int k = 0; k < KIN; k += 4) {
        int ka = k + 2 * half;
        v2f a; a.x = arow[ka]; a.y = arow[ka + 1];
        const float* wp = w + (size_t)ka * HC + col0;
        v2f b0; b0.x = wp[0];  b0.y = wp[HC];
        v2f b1; b1.x = wp[16]; b1.y = wp[HC + 16];
        acc0 = __builtin_amdgcn_wmma_f32_16x16x4_f32(false, a, false, b0,
                                                     (short)0, acc0, false, false);
        acc1 = __builtin_amdgcn_wmma_f32_16x16x4_f32(false, a, false, b1,
                                                     (short)0, acc1, false, false);
    }
    int rbase = mTile * 16 + 8 * half;
    if (rbase + 8 <= n) {                   // fast path: full tile, no per-row branches
        #pragma unroll
        for (int r = 0; r < 8; ++r) {
            float* hp = h + (size_t)(rbase + r) * HC + col0;
            hp[0]  = acc0[r];
            hp[16] = acc1[r];
        }
    } else {
        for (int r = 0; r < 8; ++r) {
            if (rbase + r < n) {
                float* hp = h + (size_t)(rbase + r) * HC + col0;
                hp[0]  = acc0[r];
                hp[16] = acc1[r];
            }
        }
    }
}

// ---------- attention logits: a_src/a_dst [N,H] (one wave per node) ----------
__global__ __launch_bounds__(256) void att_kernel(const float* __restrict__ h,
                                                  const float* __restrict__ wsrc,
                                                  const float* __restrict__ wdst,
                                                  float* __restrict__ asrc,
                                                  float* __restrict__ adst, int n) {
    int node = ((int)blockIdx.x * blockDim.x + threadIdx.x) >> 5;
    int lane = threadIdx.x & 31;
    if (node >= n) return;
    int hh = lane >> 3, c0 = lane & 7;
    const float* hp = h + (size_t)node * HC + hh * CCH + c0;
    const float* ws = wsrc + hh * CCH + c0;
    const float* wd = wdst + hh * CCH + c0;
    float ss = 0.f, sd = 0.f;
    #pragma unroll
    for (int i = 0; i < 8; ++i) {
        float v = hp[i * 8];
        ss += v * ws[i * 8];
        sd += v * wd[i * 8];
    }
    #pragma unroll
    for (int off = 4; off >= 1; off >>= 1) {
        ss += __shfl_xor(ss, off, 32);
        sd += __shfl_xor(sd, off, 32);
    }
    if (c0 == 0) {
        asrc[(size_t)node * NH + hh] = ss;
        adst[(size_t)node * NH + hh] = sd;
    }
}

// ---------- edge pass 1: segment max over dst ----------
__global__ void edge_max(const int* __restrict__ ei, const float* __restrict__ asrc,
                         const float* __restrict__ adst, unsigned* __restrict__ emax,
                         int E, int n) {
    int e = (int)blockIdx.x * blockDim.x + threadIdx.x;
    if (e >= E + n) return;
    int s, d;
    if (e < E) { s = ei[e]; d = ei[E + e]; } else { s = d = e - E; }
    #pragma unroll
    for (int hh = 0; hh < NH; ++hh) {
        float v = asrc[s * NH + hh] + adst[d * NH + hh];
        v = v > 0.f ? v : 0.2f * v;
        atomicMax(&emax[d * NH + hh], fenc(v));
    }
}

// ---------- edge pass 2: exp-sum (softmax denominator) ----------
__global__ void edge_sum(const int* __restrict__ ei, const float* __restrict__ asrc,
                         const float* __restrict__ adst, const unsigned* __restrict__ emax,
                         float* __restrict__ denom, int E, int n) {
    int e = (int)blockIdx.x * blockDim.x + threadIdx.x;
    if (e >= E + n) return;
    int s, d;
    if (e < E) { s = ei[e]; d = ei[E + e]; } else { s = d = e - E; }
    #pragma unroll
    for (int hh = 0; hh < NH; ++hh) {
        float v = asrc[s * NH + hh] + adst[d * NH + hh];
        v = v > 0.f ? v : 0.2f * v;
        atomicAdd(&denom[d * NH + hh], __expf(v - fdec(emax[d * NH + hh])));
    }
}

// ---------- edge pass 3: weighted aggregation (one wave per edge) ----------
__global__ __launch_bounds__(256) void edge_agg(const int* __restrict__ ei,
                                                const float* __restrict__ asrc,
                                                const float* __restrict__ adst,
                                                const unsigned* __restrict__ emax,
                                                const float* __restrict__ denom,
                                                const float* __restrict__ h,
                                                float* __restrict__ agg, int E, int n) {
    int e = ((int)blockIdx.x * blockDim.x + threadIdx.x) >> 5;
    int lane = threadIdx.x & 31;
    if (e >= E + n) return;
    int s, d;
    if (e < E) { s = ei[e]; d = ei[E + e]; } else { s = d = e - E; }
    int hh = lane >> 3;   // head owning this lane's 8 channels
    float v = asrc[s * NH + hh] + adst[d * NH + hh];
    v = v > 0.f ? v : 0.2f * v;
    float alpha = __expf(v - fdec(emax[d * NH + hh])) / denom[d * NH + hh];
    const float4* hp = (const float4*)(h + (size_t)s * HC + lane * 8);  // 32B/lane -> b128
    float4 h0 = hp[0], h1 = hp[1];
    float* op = agg + (size_t)d * HC + lane * 8;
    atomicAdd(&op[0], alpha * h0.x);
    atomicAdd(&op[1], alpha * h0.y);
    atomicAdd(&op[2], alpha * h0.z);
    atomicAdd(&op[3], alpha * h0.w);
    atomicAdd(&op[4], alpha * h1.x);
    atomicAdd(&op[5], alpha * h1.y);
    atomicAdd(&op[6], alpha * h1.z);
    atomicAdd(&op[7], alpha * h1.w);
}

// ---------- bias + leaky-relu + pooled segment sums ----------
__global__ void node_post(const float* __restrict__ agg, const float* __restrict__ bias,
                          const int* __restrict__ batch, float* __restrict__ pooled,
                          float* __restrict__ cnt, int n) {
    int t = (int)blockIdx.x * blockDim.x + threadIdx.x;
    if (t >= n * HC) return;
    int node = t >> 8, ch = t & 255;
    float v = agg[t] + bias[ch];
    v = v > 0.f ? v : 0.01f * v;
    int g = batch[node];
    atomicAdd(&pooled[(size_t)g * HC + ch], v);
    if (ch == 0) atomicAdd(&cnt[g], 1.0f);
}

// ---------- GEMM2: out[G,768] = (pooled/cnt) @ fc1_w + fc1_b ----------
// One wave per 16x32 slab; mean folded into A fragment, bias in epilogue.
__global__ __launch_bounds__(32) void gemm_out(const float* __restrict__ pooled,
                                               const float* __restrict__ cnt,
                                               const float* __restrict__ fw,
                                               const float* __restrict__ fb,
                                               float* __restrict__ out, int g) {
    int mTile = blockIdx.x;
    int lane = threadIdx.x & 31;
    int mr = lane & 15, half = lane >> 4;
    int row = mTile * 16 + mr;
    int col0 = blockIdx.y * 32 + mr;
    float inv = 1.0f / fmaxf(cnt[row], 1.0f);
    const float* arow = pooled + (size_t)row * HC;
    v8f acc0 = {}, acc1 = {};
    for (int k = 0; k < HC; k += 4) {
        int ka = k + 2 * half;
        v2f a; a.x = arow[ka] * inv; a.y = arow[ka + 1] * inv;
        const float* wp = fw + (size_t)ka * NOUTD + col0;
        v2f b0; b0.x = wp[0];  b0.y = wp[NOUTD];
        v2f b1; b1.x = wp[16]; b1.y = wp[NOUTD + 16];
        acc0 = __builtin_amdgcn_wmma_f32_16x16x4_f32(false, a, false, b0,
                                                     (short)0, acc0, false, false);
        acc1 = __builtin_amdgcn_wmma_f32_16x16x4_f32(false, a, false, b1,
                                                     (short)0, acc1, false, false);
    }
    float bb0 = fb[col0], bb1 = fb[col0 + 16];
    int rbase = mTile * 16 + 8 * half;
    #pragma unroll
    for (int r = 0; r < 8; ++r) {
        float* op = out + (size_t)(rbase + r) * NOUTD + col0;
        op[0]  = acc0[r] + bb0;
        op[16] = acc1[r] + bb1;
    }
}

extern "C" void kernel_launch(void* const* d_in, const int* in_sizes, int n_in,
                              void* d_out, int out_size, void* d_ws, size_t ws_size,
                              hipStream_t stream) {
    const float* x      = (const float*)d_in[0];
    const int*   ei     = (const int*)  d_in[1];
    const int*   batch  = (const int*)  d_in[2];
    const float* lin_w  = (const float*)d_in[3];
    const float* att_s  = (const float*)d_in[4];
    const float* att_d  = (const float*)d_in[5];
    const float* bias   = (const float*)d_in[6];
    const float* fc1_w  = (const float*)d_in[7];
    const float* fc1_b  = (const float*)d_in[8];
    float* out = (float*)d_out;

    const int N = in_sizes[2];           // nodes (batch vector length)
    const int E = in_sizes[1] / 2;       // edges before self-loops
    const int G = out_size / NOUTD;      // graphs

    // workspace layout (floats)
    float* ws = (float*)d_ws;
    float*    h      = ws;                                 // N*256
    float*    asrc   = h + (size_t)N * HC;                 // N*4
    float*    adst   = asrc + (size_t)N * NH;              // N*4
    unsigned* emax   = (unsigned*)(adst + (size_t)N * NH); // N*4
    float*    denom  = (float*)(emax + (size_t)N * NH);    // N*4
    float*    agg    = denom + (size_t)N * NH;             // N*256
    float*    pooled = agg + (size_t)N * HC;               // G*256
    float*    cnt    = pooled + (size_t)G * HC;            // G

    // 1. init accumulators
    init_ws<<<2048, 256, 0, stream>>>(emax, denom, agg, pooled, cnt, N, G);

    // 2. h = x @ lin_w  (WMMA f32, 16x32 slab per wave)
    dim3 g1((N + 15) / 16, HC / 32);
    gemm_x_linw<<<g1, 32, 0, stream>>>(x, lin_w, h, N);

    // 3. attention logits
    att_kernel<<<(N * 32 + 255) / 256, 256, 0, stream>>>(h, att_s, att_d, asrc, adst, N);

    // 4-6. edge passes
    int tot = E + N;
    edge_max<<<(tot + 255) / 256, 256, 0, stream>>>(ei, asrc, adst, emax, E, N);
    edge_sum<<<(tot + 255) / 256, 256, 0, stream>>>(ei, asrc, adst, emax, denom, E, N);
    edge_agg<<<((size_t)tot * 32 + 255) / 256, 256, 0, stream>>>(ei, asrc, adst, emax,
                                                                 denom, h, agg, E, N);

    // 7. bias + act + pooled sums
    node_post<<<((size_t)N * HC + 255) / 256, 256, 0, stream>>>(agg, bias, batch,
                                                                pooled, cnt, N);

    // 8. final FC (WMMA f32, 16x32 slab per wave)
    dim3 g2(G / 16, NOUTD / 32);
    gemm_out<<<g2, 32, 0, stream>>>(pooled, cnt, fc1_w, fc1_b, out, G);
}